// FocalLoss_32031866094055
// MI455X (gfx1250) — compile-verified
//
#include <hip/hip_runtime.h>
#include <hip/hip_bf16.h>
#include <math.h>

// ---------------- problem constants ----------------
#define F_   16          // B*S frames
#define A_   24000
#define C_   80
#define M_   20
#define CE_  7
#define POS_TH 0.5f
#define NEG_TH 0.4f
#define V0_  0.1f
#define V1_  0.2f
#define SL1B (1.0f/9.0f)
#define EPS_ 1e-7f
#define LN2_ 0.69314718056f

// ---------------- tiling ----------------
#define TILE  64                       // anchors per tile
#define TELEMS (TILE*C_)               // 5120 floats = 20480 bytes
#define TPF   (A_/TILE)                // 375 tiles per frame (exact)
#define TTOT  (F_*TPF)                 // 6000 tiles
#define NBLK  600                      // persistent blocks -> 10 tiles each

typedef __attribute__((ext_vector_type(2))) float v2f;
typedef __attribute__((ext_vector_type(8))) float v8f;
typedef __attribute__((ext_vector_type(4))) int   v4i;

#define AS1 __attribute__((address_space(1)))
#define AS3 __attribute__((address_space(3)))

// ---------- CDNA5 async global->LDS copy (ASYNCcnt-tracked) ----------
#if __has_builtin(__builtin_amdgcn_global_load_async_to_lds_b128)
#define HAVE_ASYNC 1
__device__ __forceinline__ void async_b128(void* lds, const void* g) {
  __builtin_amdgcn_global_load_async_to_lds_b128((AS1 v4i*)g, (AS3 v4i*)lds, 0, 0);
}
#else
#define HAVE_ASYNC 0
__device__ __forceinline__ void async_b128(void* lds, const void* g) {
  unsigned loff = (unsigned)(unsigned long long)(AS3 void*)lds;
  unsigned long long ga = (unsigned long long)(size_t)g;
  asm volatile("global_load_async_to_lds_b128 %0, %1, off" :: "v"(loff), "v"(ga) : "memory");
}
#endif

#if __has_builtin(__builtin_amdgcn_s_wait_asynccnt)
#define WAIT_ASYNC(n) __builtin_amdgcn_s_wait_asynccnt(n)
#else
#define WAIT_ASYNC(n) asm volatile("s_wait_asynccnt %0" :: "i"(n) : "memory")
#endif

#if __has_builtin(__builtin_amdgcn_wmma_f32_16x16x4_f32)
#define HAVE_WMMA4 1
#else
#define HAVE_WMMA4 0
#endif

// fast sigmoid: v_exp + v_rcp (trans pipe), ~1 ulp — fine for a loss
__device__ __forceinline__ float fast_sigmoid(float x) {
  return __builtin_amdgcn_rcpf(1.f + __expf(-x));
}

// background (y==0) focal term = 1e-4 * p * (-log(1-clip(p))).
// Argument of the log is clamped into [EPS, 1-EPS] (normal, finite), so the raw
// v_log_f32 (base-2) is safe; fold ln2*1e-4 into one constant.
// Yields exactly 0 for overwritten ignore rows (x=-1e30 -> p=0).
__device__ __forceinline__ float bg_term(float x) {
  float p  = fast_sigmoid(x);
  float q  = fminf(fmaxf(1.f - p, EPS_), 1.f - EPS_);   // = 1 - clip(p)
  float l2 = __builtin_amdgcn_logf(q);                  // log2(q) <= 0
  return p * l2 * (-(1.0f - 0.9999f) * LN2_);
}

// ---------------- helpers ----------------
__device__ __forceinline__ float block_reduce256(float v, float* red) {
  int tid = threadIdx.x;
  red[tid] = v; __syncthreads();
#pragma unroll
  for (int s = 128; s > 0; s >>= 1) {
    if (tid < s) red[tid] += red[tid + s];
    __syncthreads();
  }
  float r = red[0]; __syncthreads();
  return r;
}

// ---------------- kernel: class-balanced weights ----------------
__global__ void __launch_bounds__(256) weights_kernel(const int* __restrict__ cls,
                                                      float* __restrict__ w) {
  __shared__ float red[256];
  int tid = threadIdx.x;
  float raw = 0.f;
  if (tid < C_) {
    float n = (float)cls[tid];
    raw = (1.0f - 0.9999f) / (1.0f - powf(0.9999f, n));
  }
  float tot = block_reduce256(raw, red);
  if (tid < C_) w[tid] = raw / tot;
}

// ---------------- kernel: best anchor per GT (argmax over A, first-max) ----------------
__global__ void __launch_bounds__(256) bestprior_kernel(const float* __restrict__ gtb,
                                                        const int* __restrict__ counts,
                                                        const float* __restrict__ anchors,
                                                        int* __restrict__ bp) {
  __shared__ float sG[M_ * 4];
  __shared__ int sCnt;
  __shared__ float rv[256];
  __shared__ int   ri[256];
  int f = blockIdx.x, tid = threadIdx.x;
  if (tid < M_ * 4) sG[tid] = gtb[f * M_ * 4 + tid];
  if (tid == 0) sCnt = counts[f];
  __syncthreads();
  int cnt = sCnt;
  float bv[M_]; int bi[M_];
#pragma unroll
  for (int j = 0; j < M_; ++j) { bv[j] = -1.0f; bi[j] = A_; }
  for (int a = tid; a < A_; a += 256) {
    float acx = anchors[a*4+0], acy = anchors[a*4+1], aw = anchors[a*4+2], ah = anchors[a*4+3];
    float ax1 = acx - 0.5f*aw, ay1 = acy - 0.5f*ah, ax2 = acx + 0.5f*aw, ay2 = acy + 0.5f*ah;
    float aar = (ax2-ax1)*(ay2-ay1);
#pragma unroll
    for (int j = 0; j < M_; ++j) {
      if (j < cnt) {
        float x1=sG[j*4], y1=sG[j*4+1], x2=sG[j*4+2], y2=sG[j*4+3];
        float lx=fmaxf(x1,ax1), ly=fmaxf(y1,ay1);
        float rx=fminf(x2,ax2), ry=fminf(y2,ay2);
        float iw=fmaxf(rx-lx,0.f), ih=fmaxf(ry-ly,0.f);
        float inter=iw*ih;
        float ga=(x2-x1)*(y2-y1);
        float v = inter/(ga+aar-inter+1e-10f);
        if (v > bv[j]) { bv[j]=v; bi[j]=a; }   // strict > keeps first max
      }
    }
  }
  for (int j = 0; j < M_; ++j) {
    rv[tid]=bv[j]; ri[tid]=bi[j]; __syncthreads();
#pragma unroll
    for (int s = 128; s > 0; s >>= 1) {
      if (tid < s) {
        bool take = (rv[tid+s] > rv[tid]) || (rv[tid+s] == rv[tid] && ri[tid+s] < ri[tid]);
        if (take) { rv[tid]=rv[tid+s]; ri[tid]=ri[tid+s]; }
      }
      __syncthreads();
    }
    if (tid == 0) bp[f*M_+j] = (j < cnt) ? ri[0] : A_;
    __syncthreads();
  }
}

// ---------------- kernel: per-anchor match -> conf code {-1,0,1..M} ----------------
__global__ void __launch_bounds__(256) match_kernel(const float* __restrict__ gtb,
                                                    const int* __restrict__ counts,
                                                    const float* __restrict__ anchors,
                                                    const int* __restrict__ bp,
                                                    int* __restrict__ confOut) {
  __shared__ float sG[M_ * 4];
  __shared__ int sBP[M_];
  __shared__ int sCnt;
  int f = blockIdx.y;
  int tid = threadIdx.x;
  int a = blockIdx.x * 256 + tid;
  if (tid < M_ * 4) sG[tid] = gtb[f * M_ * 4 + tid];
  if (tid < M_)     sBP[tid] = bp[f * M_ + tid];
  if (tid == 0)     sCnt = counts[f];
  __syncthreads();
  if (a >= A_) return;
  float acx = anchors[a*4+0], acy = anchors[a*4+1], aw = anchors[a*4+2], ah = anchors[a*4+3];
  float ax1 = acx - 0.5f*aw, ay1 = acy - 0.5f*ah, ax2 = acx + 0.5f*aw, ay2 = acy + 0.5f*ah;
  float aar = (ax2-ax1)*(ay2-ay1);
  int cnt = sCnt;
  float best = -1.0f; int bidx = 0;
#pragma unroll
  for (int j = 0; j < M_; ++j) {
    float v = -1.0f;
    if (j < cnt) {
      float x1=sG[j*4], y1=sG[j*4+1], x2=sG[j*4+2], y2=sG[j*4+3];
      float lx=fmaxf(x1,ax1), ly=fmaxf(y1,ay1);
      float rx=fminf(x2,ax2), ry=fminf(y2,ay2);
      float iw=fmaxf(rx-lx,0.f), ih=fmaxf(ry-ly,0.f);
      float inter=iw*ih;
      float ga=(x2-x1)*(y2-y1);
      v = inter/(ga+aar-inter+1e-10f);
    }
    if (v > best) { best=v; bidx=j; }
  }
  int forced = -1;
  for (int j = 0; j < cnt; ++j) if (sBP[j] == a) forced = j;  // last-wins
  int conf;
  if (cnt == 0)            conf = -1;
  else if (forced >= 0)    conf = forced + 1;     // iou forced to 2.0
  else if (best < NEG_TH)  conf = 0;
  else if (best < POS_TH)  conf = -1;
  else                     conf = bidx + 1;
  confOut[f * A_ + a] = conf;
}

// ---------------- main streaming loss kernel ----------------
__device__ __forceinline__ void issue_tile(float* lbase, const float* gbase, int tid) {
#pragma unroll
  for (int s = 0; s < 5; ++s) {             // 5 * 256 * 16B = 20480B = full tile
    int off = s * 1024 + tid * 4;           // float offsets
    async_b128(lbase + off, gbase + off);
  }
}

__global__ void __launch_bounds__(256) main_loss_kernel(
    const float* __restrict__ conf, const float* __restrict__ ploc,
    const float* __restrict__ gtb,  const float* __restrict__ gtl,
    const float* __restrict__ anchors, const int* __restrict__ confT,
    const float* __restrict__ wCls, float* __restrict__ partials) {
  __shared__ __align__(16) float buf[2][TELEMS];   // 40 KB double buffer
  __shared__ float sW[C_];
  __shared__ float red[256];
  const int tid  = threadIdx.x;
  const int wave = tid >> 5;
  const int lane = tid & 31;
  if (tid < C_) sW[tid] = wCls[tid];

  float regAcc = 0.f, posAcc = 0.f, corrAcc = 0.f;
#if HAVE_WMMA4
  v8f acc = {0.f,0.f,0.f,0.f,0.f,0.f,0.f,0.f};
#else
  float clsScalar = 0.f;
#endif

  if (blockIdx.x < TTOT)
    issue_tile(&buf[0][0], conf + (size_t)blockIdx.x * TELEMS, tid);

  int it2 = 0;
  for (int t = blockIdx.x; t < TTOT; t += gridDim.x, ++it2) {
    const int cur = it2 & 1;
    const int tn  = t + gridDim.x;
    if (tn < TTOT) {
      issue_tile(&buf[cur ^ 1][0], conf + (size_t)tn * TELEMS, tid);
      WAIT_ASYNC(5);    // leave only the just-issued prefetch outstanding
    } else {
      WAIT_ASYNC(0);
    }
    const int f  = t / TPF;
    const int a0 = (t - f * TPF) * TILE;
    int code = -2;
    if (tid < TILE) code = confT[f * A_ + a0 + tid];
    // barrier + block-wide "any non-ignored anchor" flag (uniform branch later)
    int any = __syncthreads_or(code > -1);

    // ---- per-anchor phase: regression + positive corrections + ignore masking ----
    if (tid < TILE) {
      float* row = &buf[cur][tid * C_];
      if (code > 0) {
        // regression (smooth-L1 of encoded offsets)
        int a = a0 + tid;
        const float* gb = gtb + ((size_t)f * M_ + (code - 1)) * 4;
        float x1=gb[0], y1=gb[1], x2=gb[2], y2=gb[3];
        const float* an = anchors + (size_t)a * 4;
        float acx=an[0], acy=an[1], aw=an[2], ah=an[3];
        float tg[4];
        tg[0] = ((x1 + x2) * 0.5f - acx) / (V0_ * aw);
        tg[1] = ((y1 + y2) * 0.5f - acy) / (V0_ * ah);
        tg[2] = __logf(fmaxf(x2 - x1, 1e-6f) / aw) / V1_;
        tg[3] = __logf(fmaxf(y2 - y1, 1e-6f) / ah) / V1_;
        const float* pl = ploc + ((size_t)f * A_ + a) * 4;
#pragma unroll
        for (int k = 0; k < 4; ++k) {
          float n = fabsf(pl[k] - tg[k]);
          regAcc += (n < SL1B) ? (0.5f * n * n / SL1B) : (n - 0.5f * SL1B);
        }
        posAcc += 1.f;
        // classification correction for y==1 classes of this positive anchor:
        // replace the background term (added by the element loop) with the true term
        const float* yrow = gtl + ((size_t)f * M_ + (code - 1)) * C_;
        for (int c = 0; c < C_; ++c) {
          float y = yrow[c];
          if (y != 0.f) {
            float x  = row[c];
            float p  = fast_sigmoid(x);
            float pc = fminf(fmaxf(p, EPS_), 1.f - EPS_);
            float truet = (-LN2_ * __builtin_amdgcn_logf(pc)) * sW[c] * (1.f - p);
            corrAcc += truet - bg_term(x);
          }
        }
      } else if (code == -1) {
        // ignore anchor: force p=0 so the branchless element loop yields exactly 0
#pragma unroll 4
        for (int c = 0; c < C_; ++c) row[c] = -1e30f;
      }
    }
    __syncthreads();

    // ---- branchless background sweep: 5120 elems, 640/wave, matrix-pipe reduce ----
    if (any) {
      const v2f* bf2 = (const v2f*)&buf[cur][0];
      const int base2 = wave * (TELEMS / 16) + lane;   // v2f index
      for (int itw = 0; itw < (TELEMS / 8) / 64; ++itw) {
        v2f xv = bf2[base2 + itw * 32];
        float v0 = bg_term(xv.x);
        float v1 = bg_term(xv.y);
#if HAVE_WMMA4
        v2f aF; aF.x = v0; aF.y = v1;
        v2f bF; bF.x = 1.0f; bF.y = 1.0f;
        // D = A(16x4) * ones(4x16) + C  => every C column accumulates the full sum
        acc = __builtin_amdgcn_wmma_f32_16x16x4_f32(false, aF, false, bF,
                                                    (short)0, acc, false, false);
#else
        clsScalar += v0 + v1;
#endif
      }
    }
    __syncthreads();
  }

  float clsVal;
#if HAVE_WMMA4
  clsVal = acc[0]+acc[1]+acc[2]+acc[3]+acc[4]+acc[5]+acc[6]+acc[7]
         + 16.f * corrAcc;                 // fold corrections into the /16 norm
#else
  clsVal = clsScalar + corrAcc;
#endif
  float cls = block_reduce256(clsVal, red);
  float rg  = block_reduce256(regAcc, red);
  float np  = block_reduce256(posAcc, red);
  if (tid == 0) {
    partials[blockIdx.x * 4 + 0] = rg;
#if HAVE_WMMA4
    partials[blockIdx.x * 4 + 1] = cls * (1.0f / 16.0f);  // sum over C = 16 * total
#else
    partials[blockIdx.x * 4 + 1] = cls;
#endif
    partials[blockIdx.x * 4 + 2] = np;
  }
}

// ---------------- ego focal loss (tiny) ----------------
__global__ void __launch_bounds__(256) ego_kernel(const float* __restrict__ ep,
                                                  const int* __restrict__ el,
                                                  float* __restrict__ out) {
  __shared__ float col[CE_];
  __shared__ float red[256];
  int tid = threadIdx.x;
  if (tid < CE_) {
    float any = 0.f;
    for (int i = 0; i < F_; ++i) { int l = el[i]; if (l > -1 && l == tid) any = 1.f; }
    col[tid] = any;
  }
  __syncthreads();
  float sum = 0.f;
  for (int e = tid; e < F_ * CE_; e += 256) {
    int bs = e / CE_; int c = e - bs * CE_;
    if (el[bs] > -1) {
      float p  = 1.f / (1.f + __expf(-ep[e]));
      float oh = col[c];
      float pc = fminf(fmaxf(p, EPS_), 1.f - EPS_);
      float bce = -(oh * __logf(pc) + (1.f - oh) * __logf(1.f - pc));
      float af  = 0.25f * oh + 0.75f * (1.f - oh);
      float u   = 1.f - (p * oh + (1.f - p) * (1.f - oh));
      sum += bce * af * u * u;
    }
  }
  float nv = (tid < F_ && el[tid] > -1) ? 1.f : 0.f;
  float stot = block_reduce256(sum, red);
  float ntot = block_reduce256(nv,  red);
  if (tid == 0) out[0] = (ntot > 0.f) ? stot / fmaxf(ntot, 1.f) : 0.f;
}

// ---------------- finalize ----------------
__global__ void __launch_bounds__(256) finalize_kernel(const float* __restrict__ part,
                                                       const float* __restrict__ ego,
                                                       float* __restrict__ out) {
  __shared__ float red[256];
  int tid = threadIdx.x;
  float a = 0.f, b = 0.f, c = 0.f;
  for (int i = tid; i < NBLK; i += 256) {
    a += part[i * 4 + 0];
    b += part[i * 4 + 1];
    c += part[i * 4 + 2];
  }
  a = block_reduce256(a, red);
  b = block_reduce256(b, red);
  c = block_reduce256(c, red);
  if (tid == 0) {
    float np = fmaxf(1.f, c);
    out[0] = a / (np * 4.f);
    out[1] = (b / np) * 0.125f + ego[0] * 0.25f;
  }
}

// ---------------- host launcher ----------------
extern "C" void kernel_launch(void* const* d_in, const int* in_sizes, int n_in,
                              void* d_out, int out_size, void* d_ws, size_t ws_size,
                              hipStream_t stream) {
  (void)in_sizes; (void)n_in; (void)out_size; (void)ws_size;
  const float* confidence = (const float*)d_in[0];
  const float* ploc       = (const float*)d_in[1];
  const float* gtb        = (const float*)d_in[2];
  const float* gtl        = (const float*)d_in[3];
  const int*   counts     = (const int*)  d_in[4];
  const float* anchors    = (const float*)d_in[5];
  const float* ego_preds  = (const float*)d_in[6];
  const int*   ego_labels = (const int*)  d_in[7];
  const int*   cls_num    = (const int*)  d_in[8];
  float* out = (float*)d_out;

  // workspace layout (~1.55 MB total)
  int*   ws_conf = (int*)d_ws;                 // F_*A_
  int*   ws_bp   = ws_conf + F_ * A_;          // F_*M_
  float* ws_w    = (float*)(ws_bp + F_ * M_);  // C_
  float* ws_ego  = ws_w + C_;                  // 1 (+3 pad)
  float* ws_part = ws_ego + 4;                 // NBLK*4

  weights_kernel<<<1, 256, 0, stream>>>(cls_num, ws_w);
  bestprior_kernel<<<F_, 256, 0, stream>>>(gtb, counts, anchors, ws_bp);
  dim3 mg((A_ + 255) / 256, F_);
  match_kernel<<<mg, 256, 0, stream>>>(gtb, counts, anchors, ws_bp, ws_conf);
  main_loss_kernel<<<NBLK, 256, 0, stream>>>(confidence, ploc, gtb, gtl, anchors,
                                             ws_conf, ws_w, ws_part);
  ego_kernel<<<1, 256, 0, stream>>>(ego_preds, ego_labels, ws_ego);
  finalize_kernel<<<1, 256, 0, stream>>>(ws_part, ws_ego, out);
}